// ModelB_A100_Loss_58007828300468
// MI455X (gfx1250) — compile-verified
//
#include <hip/hip_runtime.h>
#include <hip/hip_bf16.h>
#include <stdint.h>

#define B_ 64
#define N_ 512
#define C_ 2
#define EPS_ 1e-8f
#define ROWS_PER_BLOCK 8

typedef __attribute__((ext_vector_type(2))) float v2f;
typedef __attribute__((ext_vector_type(4))) float v4f;
typedef __attribute__((ext_vector_type(8))) float v8f;

// workspace layout (floats)
#define WS_EDGE    0
#define WS_SIM     1
#define WS_MSE     2
#define WS_SMOOTH  3
#define WS_CNTLOSS 4
#define WS_MASKTOT 5
#define WS_DOT     16
#define WS_NA2     (16 + 64)
#define WS_NT2     (16 + 128)
#define WS_ENT     (16 + 192)
#define WS_CON     (16 + 256)
#define WS_CNT     (16 + 320)
#define WS_TOTAL   400

__global__ void zero_ws_kernel(float* ws) {
    int t = threadIdx.x;
    if (t < WS_TOTAL) ws[t] = 0.0f;
}

__device__ inline float wave_sum32(float v) {
#pragma unroll
    for (int o = 16; o > 0; o >>= 1) v += __shfl_xor(v, o, 32);
    return v;
}

// Per-batch: count = popcount(mask prefix), coord MSE + smooth-L1, count huber loss.
__global__ void prep_kernel(const float* __restrict__ pc,
                            const float* __restrict__ pts,
                            const float* __restrict__ ncounts,
                            const unsigned char* __restrict__ masks,
                            float* __restrict__ ws) {
    int b = blockIdx.x;
    int t = threadIdx.x;                       // 0..511
    bool m = masks[b * N_ + t] != 0;
    float mse = 0.0f, smooth = 0.0f, cnt = m ? 1.0f : 0.0f;
    if (m) {
        float d0 = pc[((size_t)b * N_ + t) * C_ + 0] - pts[((size_t)b * N_ + t) * C_ + 0];
        float d1 = pc[((size_t)b * N_ + t) * C_ + 1] - pts[((size_t)b * N_ + t) * C_ + 1];
        mse = d0 * d0 + d1 * d1;
        float a0 = fabsf(d0), a1 = fabsf(d1);
        smooth = (a0 < 1.0f ? 0.5f * d0 * d0 : a0 - 0.5f)
               + (a1 < 1.0f ? 0.5f * d1 * d1 : a1 - 0.5f);
    }
    __shared__ float s_mse[16], s_sm[16], s_cnt[16];
    int wave = t >> 5, lane = t & 31;
    float wm = wave_sum32(mse), wsm = wave_sum32(smooth), wc = wave_sum32(cnt);
    if (lane == 0) { s_mse[wave] = wm; s_sm[wave] = wsm; s_cnt[wave] = wc; }
    __syncthreads();
    if (t == 0) {
        float tm = 0.0f, tsm = 0.0f, tc = 0.0f;
        for (int i = 0; i < 16; ++i) { tm += s_mse[i]; tsm += s_sm[i]; tc += s_cnt[i]; }
        atomicAdd(&ws[WS_MSE], tm);
        atomicAdd(&ws[WS_SMOOTH], tsm);
        atomicAdd(&ws[WS_MASKTOT], tc);
        ws[WS_CNT + b] = tc;
        float dc = ncounts[b] - tc;
        float adc = fabsf(dc);
        float h = (adc <= 1.0f) ? 0.5f * dc * dc : adc - 0.5f;
        atomicAdd(&ws[WS_CNTLOSS], h * (1.0f / (float)B_));
    }
}

struct Acc {
    float edge, sim, dot, na2, nt2, ent, con;
};

__device__ inline void acc_elem(Acc& a, float p, float t0, float s) {
    float lp  = __logf(p);
    float l1p = __logf(1.0f - p);
    float ts  = 0.9f * t0 + 0.05f;
    a.edge += ts * lp + (1.0f - ts) * l1p;
    float dsv = s - t0;
    a.sim  += dsv * dsv;
    a.dot  += p * t0;
    a.na2  += p * p;
    a.nt2  += t0 * t0;
    a.ent  += p * __logf(p + EPS_) + (1.0f - p) * __logf(1.0f - p + EPS_);
    a.con  += fabsf(p - 0.5f);
}

// Fused pass over the valid count x count prefix block of the three big tensors.
// One wave per row. Full 128-column chunks use non-temporal B128 vector loads
// (streaming data, zero reuse -> TH=NT keeps it out of L2); ragged tail is scalar.
__global__ void __launch_bounds__(256)
main_kernel(const float* __restrict__ p_,   // adjacency_matrix
            const float* __restrict__ s_,   // raw_similarity
            const float* __restrict__ t_,   // adjacency
            float* __restrict__ ws) {
    int b = blockIdx.y;
    int count = (int)ws[WS_CNT + b];
    int row0 = blockIdx.x * ROWS_PER_BLOCK;
    if (row0 >= count) return;                 // uniform per block (before any barrier)

    int wave = threadIdx.x >> 5, lane = threadIdx.x & 31;
    int row = row0 + wave;
    Acc a = {0.0f, 0.0f, 0.0f, 0.0f, 0.0f, 0.0f, 0.0f};

    if (row < count) {
        size_t base = ((size_t)b * N_ + row) * N_;
        int jv = 0;
        // vectorized body: 128 columns per iteration (lane*4, 16B-aligned)
        for (; jv + 128 <= count; jv += 128) {
            size_t off = base + jv + lane * 4;
            __builtin_prefetch(&p_[off + 128], 0, 0);   // global_prefetch_b8
            __builtin_prefetch(&t_[off + 128], 0, 0);
            __builtin_prefetch(&s_[off + 128], 0, 0);
            v4f pv = __builtin_nontemporal_load((const v4f*)(p_ + off));
            v4f tv = __builtin_nontemporal_load((const v4f*)(t_ + off));
            v4f sv = __builtin_nontemporal_load((const v4f*)(s_ + off));
#pragma unroll
            for (int k = 0; k < 4; ++k) acc_elem(a, pv[k], tv[k], sv[k]);
        }
        // ragged tail: scalar, lanes stride over remaining columns
        for (int j = jv + lane; j < count; j += 32) {
            float p  = __builtin_nontemporal_load(p_ + base + j);
            float t0 = __builtin_nontemporal_load(t_ + base + j);
            float s  = __builtin_nontemporal_load(s_ + base + j);
            acc_elem(a, p, t0, s);
        }
    }

    a.edge = wave_sum32(a.edge); a.sim = wave_sum32(a.sim);
    a.dot  = wave_sum32(a.dot);  a.na2 = wave_sum32(a.na2);
    a.nt2  = wave_sum32(a.nt2);  a.ent = wave_sum32(a.ent);
    a.con  = wave_sum32(a.con);

    __shared__ float s_acc[7][ROWS_PER_BLOCK];
    if (lane == 0) {
        s_acc[0][wave] = a.edge; s_acc[1][wave] = a.sim; s_acc[2][wave] = a.dot;
        s_acc[3][wave] = a.na2;  s_acc[4][wave] = a.nt2; s_acc[5][wave] = a.ent;
        s_acc[6][wave] = a.con;
    }
    __syncthreads();
    if (threadIdx.x == 0) {
        float r[7];
        for (int q = 0; q < 7; ++q) {
            float v = 0.0f;
            for (int w = 0; w < ROWS_PER_BLOCK; ++w) v += s_acc[q][w];
            r[q] = v;
        }
        atomicAdd(&ws[WS_EDGE], r[0]);
        atomicAdd(&ws[WS_SIM],  r[1]);
        atomicAdd(&ws[WS_DOT + b], r[2]);
        atomicAdd(&ws[WS_NA2 + b], r[3]);
        atomicAdd(&ws[WS_NT2 + b], r[4]);
        atomicAdd(&ws[WS_ENT + b], r[5]);
        atomicAdd(&ws[WS_CON + b], r[6]);
    }
}

// One wave. Per-lane: batches (lane, lane+32). WMMA f32 16x16x4 with a ones
// B-matrix row-sums the 16x4 A-matrix (= all 64 per-batch values) in one op.
__global__ void finalize_kernel(const float* __restrict__ temp,
                                const float* __restrict__ resw,
                                const float* __restrict__ ws,
                                float* __restrict__ out) {
    int lane = threadIdx.x;                    // 32 threads, full EXEC (WMMA requirement)
    float contrib[2], c2p[2];
#pragma unroll
    for (int h = 0; h < 2; ++h) {
        int b = lane + h * 32;
        float cnt  = ws[WS_CNT + b];
        float dotv = ws[WS_DOT + b];
        float na   = sqrtf(ws[WS_NA2 + b]);
        float nt   = sqrtf(ws[WS_NT2 + b]);
        float cosv = dotv / (fmaxf(na, EPS_) * fmaxf(nt, EPS_));
        float n2   = fmaxf(cnt * cnt, 1.0f);
        float cond = (cnt > 5.0f && cnt <= 50.0f) ? 1.0f : 0.0f;
        float entb = -ws[WS_ENT + b] / n2;
        float conb =  ws[WS_CON + b] / n2;
        contrib[h] = cond * (-cosv - 0.2f * conb + 0.1f * entb);
        c2p[h]     = cnt * cnt;
    }

    float ari_total, cnt2;
#if __has_builtin(__builtin_amdgcn_wmma_f32_16x16x4_f32)
    // A(16x4) layout: lanes 0-15 VGPR0->K0, VGPR1->K1; lanes 16-31 VGPR0->K2, VGPR1->K3.
    // Lane L carries batches L and L+32: all 64 covered exactly once.
    v2f a1; a1.x = contrib[0]; a1.y = contrib[1];
    v2f a2; a2.x = c2p[0];     a2.y = c2p[1];
    v2f ones; ones.x = 1.0f; ones.y = 1.0f;
    v8f z = {};
    v8f d1 = __builtin_amdgcn_wmma_f32_16x16x4_f32(false, a1, false, ones, (short)0, z, false, false);
    v8f d2 = __builtin_amdgcn_wmma_f32_16x16x4_f32(false, a2, false, ones, (short)0, z, false, false);
    float s1 = 0.0f, s2 = 0.0f;
#pragma unroll
    for (int i = 0; i < 8; ++i) { s1 += d1[i]; s2 += d2[i]; }
    // lanes 0-15 hold rows 0-7, lanes 16-31 hold rows 8-15 -> one xor-16 finishes it
    ari_total = s1 + __shfl_xor(s1, 16, 32);
    cnt2      = s2 + __shfl_xor(s2, 16, 32);
#else
    float s1 = contrib[0] + contrib[1];
    float s2 = c2p[0] + c2p[1];
    ari_total = wave_sum32(s1);
    cnt2      = wave_sum32(s2);
#endif

    if (lane == 0) {
        float cnt2m     = fmaxf(cnt2, 1.0f);
        float edge_loss = -ws[WS_EDGE] / cnt2m;
        float sim_loss  =  ws[WS_SIM]  / cnt2m;
        float cnt_coord = fmaxf(ws[WS_MASKTOT] * (float)C_, 1.0f);
        float coord_loss = 0.7f * (ws[WS_MSE] / cnt_coord) + 0.3f * (ws[WS_SMOOTH] / cnt_coord);
        float count_loss = ws[WS_CNTLOSS];
        float treg = fabsf(temp[0] - 1.0f);
        float rreg = fabsf(resw[0] - 0.5f);
        float total = coord_loss + 2.0f * edge_loss + 0.1f * count_loss
                    + 0.3f * sim_loss + 0.01f * (treg + rreg) + ari_total;
        out[0] = total;
    }
}

extern "C" void kernel_launch(void* const* d_in, const int* in_sizes, int n_in,
                              void* d_out, int out_size, void* d_ws, size_t ws_size,
                              hipStream_t stream) {
    (void)in_sizes; (void)n_in; (void)out_size; (void)ws_size;
    const float* pc      = (const float*)d_in[0];          // predicted_coords (B,N,C)
    const float* adjm    = (const float*)d_in[1];          // adjacency_matrix (B,N,N)
    const float* ncounts = (const float*)d_in[2];          // node_counts (B,)
    const float* simm    = (const float*)d_in[3];          // raw_similarity (B,N,N)
    const float* temp    = (const float*)d_in[4];          // temperature
    const float* resw    = (const float*)d_in[5];          // residual_weight
    const float* pts     = (const float*)d_in[6];          // points (B,N,C)
    const float* adj     = (const float*)d_in[7];          // adjacency (B,N,N)
    const unsigned char* masks = (const unsigned char*)d_in[8]; // node_masks (B,N) bool
    float* ws  = (float*)d_ws;
    float* out = (float*)d_out;

    hipLaunchKernelGGL(zero_ws_kernel, dim3(1), dim3(512), 0, stream, ws);
    hipLaunchKernelGGL(prep_kernel, dim3(B_), dim3(N_), 0, stream, pc, pts, ncounts, masks, ws);
    hipLaunchKernelGGL(main_kernel, dim3(N_ / ROWS_PER_BLOCK, B_), dim3(256), 0, stream,
                       adjm, simm, adj, ws);
    hipLaunchKernelGGL(finalize_kernel, dim3(1), dim3(32), 0, stream, temp, resw, ws, out);
}